// Model_11845519802433
// MI455X (gfx1250) — compile-verified
//
#include <hip/hip_runtime.h>
#include <hip/hip_bf16.h>

// ---------------------------------------------------------------------------
// GNN step for MI455X (gfx1250): fp32 WMMA (v_wmma_f32_16x16x4_f32) for the
// encoder / message / update GEMMs, native u32 atomic-max (order-preserving
// float->uint key) for segment_max aggregation.
// Round 2: message kernel processes 32 edges/block with two M-tiles per wave,
// so every packed-B global_load_b64 feeds two WMMAs (halves B traffic).
// ---------------------------------------------------------------------------

#define N_NODES 50000
#define N_EDGES 800000
#define IN_DIM 16
#define HIDDEN 64

typedef float v2f __attribute__((ext_vector_type(2)));
typedef float v8f __attribute__((ext_vector_type(8)));

__device__ __forceinline__ v8f wmma_f32(v2f a, v2f b, v8f c) {
    // 8 args: (neg_a, A, neg_b, B, c_mod, C, reuse_a, reuse_b)
    return __builtin_amdgcn_wmma_f32_16x16x4_f32(false, a, false, b,
                                                 (short)0, c, false, false);
}

// Order-preserving float <-> uint key (unsigned compare == float compare).
__device__ __forceinline__ unsigned fkey(float f) {
    unsigned u = __float_as_uint(f);
    return (u & 0x80000000u) ? ~u : (u | 0x80000000u);
}
__device__ __forceinline__ float funkey(unsigned k) {
    unsigned u = (k & 0x80000000u) ? (k & 0x7FFFFFFFu) : ~k;
    return __uint_as_float(u);
}

// ---------------------------------------------------------------------------
// Pack a row-major W[K][64] (row stride 64 floats) into WMMA-B layout:
// P[nt][ks][lane] = {W[ks*4 + 2*(lane>=16)][nt*16 + lane%16],
//                    W[ks*4 + 2*(lane>=16)+1][...]}
// ---------------------------------------------------------------------------
__global__ void pack_w_kernel(const float* __restrict__ W, v2f* __restrict__ P,
                              int ksteps) {
    int idx = blockIdx.x * blockDim.x + threadIdx.x;
    int total = 4 * ksteps * 32;
    if (idx >= total) return;
    int lane = idx & 31;
    int ks   = (idx >> 5) % ksteps;
    int nt   = idx / (32 * ksteps);
    int n  = nt * 16 + (lane & 15);
    int kb = ks * 4 + ((lane >> 4) << 1);
    v2f t;
    t[0] = W[(size_t)kb * 64 + n];
    t[1] = W[(size_t)(kb + 1) * 64 + n];
    P[idx] = t;
}

__global__ void init_kernel(unsigned* __restrict__ aggK, int* __restrict__ deg,
                            int nk, int nd) {
    int i = blockIdx.x * blockDim.x + threadIdx.x;
    if (i < nk) aggK[i] = 0u;
    if (i < nd) deg[i] = 0;
}

// ---------------------------------------------------------------------------
// Encoder: z = x @ enc_W[0:16,:] + enc_b   (h0 == 0)
// 16 nodes per block, 4 waves, each wave one 16-wide N tile, K=16 (4 WMMA).
// ---------------------------------------------------------------------------
__global__ __launch_bounds__(128) void enc_kernel(
    const float* __restrict__ x, const v2f* __restrict__ Penc,
    const float* __restrict__ encb, float* __restrict__ z) {
    __shared__ __align__(16) float fx[16][16];
    int tid = threadIdx.x;
    int n0 = blockIdx.x * 16;
    if (tid < 64) {
        int m = tid >> 2, j = tid & 3;
        *(float4*)&fx[m][j * 4] =
            *(const float4*)(x + (size_t)(n0 + m) * IN_DIM + j * 4);
    }
    __syncthreads();
    int wv = tid >> 5, lane = tid & 31;
    int half = lane >> 4, l16 = lane & 15;
    v8f c = {};
    const v2f* Pw = Penc + (size_t)wv * 4 * 32;
#pragma unroll
    for (int ks = 0; ks < 4; ++ks) {
        v2f a = *(const v2f*)&fx[l16][ks * 4 + 2 * half];
        v2f b = Pw[ks * 32 + lane];
        c = wmma_f32(a, b, c);
    }
    int n = wv * 16 + l16;
    float bb = encb[n];
#pragma unroll
    for (int r = 0; r < 8; ++r) {
        int mm = r + 8 * half;
        z[(size_t)(n0 + mm) * HIDDEN + n] = c[r] + bb;
    }
}

// ---------------------------------------------------------------------------
// Messages + segment_max: per 32-edge tile, msgs = [z[s]|z[d]] @ msg_W[0:128]
// (WMMA, K=128) + w*msg_W[128,:] + msg_b, then atomic-max into agg keys.
// 4 waves; each wave owns one 16-wide N tile and carries TWO accumulators
// (edge rows 0-15 and 16-31) so each packed-B load feeds two WMMAs.
// ---------------------------------------------------------------------------
__global__ __launch_bounds__(128) void msg_kernel(
    const float* __restrict__ z, const int* __restrict__ esrc,
    const int* __restrict__ edst, const float* __restrict__ ew,
    const v2f* __restrict__ Pmsg, const float* __restrict__ msgW,
    const float* __restrict__ msgb, unsigned* __restrict__ aggK,
    int* __restrict__ deg) {
    __shared__ __align__(16) float feat[32][128];
    __shared__ float wsh[32];
    __shared__ int ssrc[32];
    int tid = threadIdx.x;
    int e0 = blockIdx.x * 32;

    if (tid < 32) {
        int e = e0 + tid;
        int s = esrc[e];
        ssrc[tid] = s;
        wsh[tid] = ew[e];
        deg[s] = 1;  // racing writes of same value are fine
    }
    // Gather [z[src] | z[dst]] -> feat[32][128]; each thread moves 32 floats.
    {
        int m = tid >> 2, part = tid & 3;
        int s = esrc[e0 + m], d = edst[e0 + m];
        const float* row = (part < 2)
                               ? z + (size_t)s * HIDDEN + part * 32
                               : z + (size_t)d * HIDDEN + (part - 2) * 32;
        float* dstp = &feat[m][part * 32];
#pragma unroll
        for (int j = 0; j < 8; ++j)
            *(float4*)(dstp + j * 4) = *(const float4*)(row + j * 4);
    }
    __syncthreads();

    int wv = tid >> 5, lane = tid & 31;
    int half = lane >> 4, l16 = lane & 15;
    v8f c0 = {};
    v8f c1 = {};
    const v2f* Pw = Pmsg + (size_t)wv * 32 * 32;  // 32 k-steps per N tile
#pragma unroll 8
    for (int ks = 0; ks < 32; ++ks) {
        v2f b  = Pw[ks * 32 + lane];
        v2f a0 = *(const v2f*)&feat[l16][ks * 4 + 2 * half];
        v2f a1 = *(const v2f*)&feat[16 + l16][ks * 4 + 2 * half];
        c0 = wmma_f32(a0, b, c0);
        c1 = wmma_f32(a1, b, c1);
    }
    int n = wv * 16 + l16;
    float wl = msgW[(size_t)128 * 64 + n];  // last row (edge-weight column)
    float bb = msgb[n];
#pragma unroll
    for (int r = 0; r < 8; ++r) {
        int mm0 = r + 8 * half;
        int mm1 = 16 + mm0;
        float v0 = c0[r] + wsh[mm0] * wl + bb;
        float v1 = c1[r] + wsh[mm1] * wl + bb;
        atomicMax(&aggK[(size_t)ssrc[mm0] * HIDDEN + n], fkey(v0));
        atomicMax(&aggK[(size_t)ssrc[mm1] * HIDDEN + n], fkey(v1));
    }
}

// ---------------------------------------------------------------------------
// Update MLP + decoder: per 16-node tile
//   t = relu([z|agg] @ W1 + b1)  (K=128 WMMA)
//   h = t @ W2 + b2              (K=64 WMMA)
//   y = [z|h] @ dec_W + dec_b
// ---------------------------------------------------------------------------
__global__ __launch_bounds__(128) void upd_kernel(
    const float* __restrict__ z, const unsigned* __restrict__ aggK,
    const int* __restrict__ deg, const v2f* __restrict__ Pu1,
    const float* __restrict__ ub1, const v2f* __restrict__ Pu2,
    const float* __restrict__ ub2, const float* __restrict__ decW,
    const float* __restrict__ decb, float* __restrict__ y,
    float* __restrict__ h) {
    __shared__ __align__(16) float feat[16][128];
    __shared__ __align__(16) float t1[16][64];
    __shared__ __align__(16) float t2[16][64];
    int tid = threadIdx.x;
    int n0 = blockIdx.x * 16;

    {
        int m = tid >> 3, part = tid & 7;
        size_t node = (size_t)(n0 + m);
        int dg = deg[node];
#pragma unroll
        for (int j = 0; j < 4; ++j) {
            int col = part * 16 + j * 4;
            float4 v;
            if (col < 64) {
                v = *(const float4*)(z + node * HIDDEN + col);
            } else if (dg) {
                int c2 = col - 64;
                v.x = funkey(aggK[node * HIDDEN + c2 + 0]);
                v.y = funkey(aggK[node * HIDDEN + c2 + 1]);
                v.z = funkey(aggK[node * HIDDEN + c2 + 2]);
                v.w = funkey(aggK[node * HIDDEN + c2 + 3]);
            } else {
                v = make_float4(0.f, 0.f, 0.f, 0.f);
            }
            *(float4*)&feat[m][col] = v;
        }
    }
    __syncthreads();

    int wv = tid >> 5, lane = tid & 31;
    int half = lane >> 4, l16 = lane & 15;
    int n = wv * 16 + l16;

    // GEMM1: K = 128
    {
        v8f c = {};
        const v2f* Pw = Pu1 + (size_t)wv * 32 * 32;
#pragma unroll 8
        for (int ks = 0; ks < 32; ++ks) {
            v2f a = *(const v2f*)&feat[l16][ks * 4 + 2 * half];
            v2f b = Pw[ks * 32 + lane];
            c = wmma_f32(a, b, c);
        }
        float bb = ub1[n];
#pragma unroll
        for (int r = 0; r < 8; ++r) {
            int mm = r + 8 * half;
            float v = c[r] + bb;
            t1[mm][n] = v > 0.f ? v : 0.f;
        }
    }
    __syncthreads();

    // GEMM2: K = 64
    {
        v8f c = {};
        const v2f* Pw = Pu2 + (size_t)wv * 16 * 32;
#pragma unroll 8
        for (int ks = 0; ks < 16; ++ks) {
            v2f a = *(const v2f*)&t1[l16][ks * 4 + 2 * half];
            v2f b = Pw[ks * 32 + lane];
            c = wmma_f32(a, b, c);
        }
        float bb = ub2[n];
#pragma unroll
        for (int r = 0; r < 8; ++r) {
            int mm = r + 8 * half;
            float v = c[r] + bb;
            t2[mm][n] = v;
            h[(size_t)(n0 + mm) * HIDDEN + n] = v;
        }
    }
    __syncthreads();

    // Decoder: y = [z|h] @ dec_W + dec_b
    if (tid < 16) {
        float acc = decb[0];
#pragma unroll 8
        for (int k = 0; k < 64; ++k) acc += feat[tid][k] * decW[k];
#pragma unroll 8
        for (int k = 0; k < 64; ++k) acc += t2[tid][k] * decW[64 + k];
        y[n0 + tid] = acc;
    }
}

// ---------------------------------------------------------------------------
extern "C" void kernel_launch(void* const* d_in, const int* in_sizes, int n_in,
                              void* d_out, int out_size, void* d_ws,
                              size_t ws_size, hipStream_t stream) {
    const float* x    = (const float*)d_in[0];
    const int*   esrc = (const int*)d_in[1];
    const int*   edst = (const int*)d_in[2];
    const float* ew   = (const float*)d_in[3];
    const float* encW = (const float*)d_in[4];
    const float* encb = (const float*)d_in[5];
    const float* msgW = (const float*)d_in[6];
    const float* msgb = (const float*)d_in[7];
    const float* uW1  = (const float*)d_in[8];
    const float* ub1  = (const float*)d_in[9];
    const float* uW2  = (const float*)d_in[10];
    const float* ub2  = (const float*)d_in[11];
    const float* decW = (const float*)d_in[12];
    const float* decb = (const float*)d_in[13];

    float* y = (float*)d_out;            // [N_NODES]
    float* h = y + N_NODES;              // [N_NODES * HIDDEN]

    // Workspace layout (256B aligned slices)
    char* ws = (char*)d_ws;
    size_t off = 0;
    auto take = [&](size_t bytes) {
        char* p = ws + off;
        off = (off + bytes + 255) & ~(size_t)255;
        return p;
    };
    unsigned* aggK = (unsigned*)take((size_t)N_NODES * HIDDEN * 4);
    int*      deg  = (int*)take((size_t)N_NODES * 4);
    float*    z    = (float*)take((size_t)N_NODES * HIDDEN * 4);
    v2f*      Penc = (v2f*)take((size_t)4 * 4 * 32 * 8);    // K=16
    v2f*      Pmsg = (v2f*)take((size_t)4 * 32 * 32 * 8);   // K=128
    v2f*      Pu1  = (v2f*)take((size_t)4 * 32 * 32 * 8);   // K=128
    v2f*      Pu2  = (v2f*)take((size_t)4 * 16 * 32 * 8);   // K=64
    (void)ws_size; (void)in_sizes; (void)n_in; (void)out_size;

    // Init aggregate keys (+deg)
    {
        int nk = N_NODES * HIDDEN, nd = N_NODES;
        int thr = 256, blk = (nk + thr - 1) / thr;
        init_kernel<<<blk, thr, 0, stream>>>(aggK, deg, nk, nd);
    }
    // Pack weights into WMMA-B layout
    {
        int thr = 256;
        pack_w_kernel<<<(4 * 4 * 32 + thr - 1) / thr, thr, 0, stream>>>(encW, Penc, 4);
        pack_w_kernel<<<(4 * 32 * 32 + thr - 1) / thr, thr, 0, stream>>>(msgW, Pmsg, 32);
        pack_w_kernel<<<(4 * 32 * 32 + thr - 1) / thr, thr, 0, stream>>>(uW1, Pu1, 32);
        pack_w_kernel<<<(4 * 16 * 32 + thr - 1) / thr, thr, 0, stream>>>(uW2, Pu2, 16);
    }
    // Encoder
    enc_kernel<<<N_NODES / 16, 128, 0, stream>>>(x, Penc, encb, z);
    // Messages + max aggregation (32 edges per block)
    msg_kernel<<<N_EDGES / 32, 128, 0, stream>>>(z, esrc, edst, ew, Pmsg, msgW,
                                                 msgb, aggK, deg);
    // Update MLP + decoder
    upd_kernel<<<N_NODES / 16, 128, 0, stream>>>(z, aggK, deg, Pu1, ub1, Pu2,
                                                 ub2, decW, decb, y, h);
}